// SoftDecisionTree_84602265797131
// MI455X (gfx1250) — compile-verified
//
#include <hip/hip_runtime.h>
#include <math.h>

typedef __attribute__((ext_vector_type(2))) float v2f;
typedef __attribute__((ext_vector_type(8))) float v8f;

#define INPUT_DIM 1024
#define BATCH     8192
#define NODES_PAD 256     // 255 real internal nodes, padded
#define NUM_NODES 255
#define DEPTH     8
#define MTILE     32      // batch rows per block
#define KC        32      // k-chunk staged in LDS

#define XS_STRIDE 36      // 32 + 4 pad floats  (bank rotate 36 % 64)
#define WS_STRIDE 36
#define LG_STRIDE 260     // 256 + 4 pad floats

#define XS_FLOATS   (MTILE * XS_STRIDE)        // 1152
#define W_FLOATS    (NODES_PAD * WS_STRIDE)    // 9216
#define GEMM_FLOATS (XS_FLOATS + W_FLOATS)     // 10368
#define LG_FLOATS   (MTILE * LG_STRIDE)        // 8320
#define SMEM_FLOATS (GEMM_FLOATS > LG_FLOATS ? GEMM_FLOATS : LG_FLOATS)

__global__ __launch_bounds__(256)
void sdt_fused_kernel(const float* __restrict__ x,      // [8192][1024]
                      const float* __restrict__ wts,    // [255][1024]
                      const float* __restrict__ bias,   // [255]
                      const float* __restrict__ leafv,  // [256]
                      float* __restrict__ out)          // [8192]
{
    __shared__ __align__(16) float smem[SMEM_FLOATS];
    float* xs = smem;               // [MTILE][XS_STRIDE]    (GEMM phase)
    float* wl = smem + XS_FLOATS;   // [NODES_PAD][WS_STRIDE](GEMM phase)
    float* lg = smem;               // [MTILE][LG_STRIDE]    (combine phase)

    const int tid  = threadIdx.x;
    const int wave = tid >> 5;
    const int lane = tid & 31;
    const int lmod = lane & 15;
    const int lhalf = lane >> 4;        // 0/1
    const int bm = blockIdx.x * MTILE;

    // wave tiling: 2 M-tiles x 4 N-quarters (each quarter = 4 N-tiles of 16)
    const int mw = wave & 1;            // M-tile 0..1
    const int nq = wave >> 1;           // N-quarter 0..3

    v8f acc[4];
    {
        v8f z8 = {0.f,0.f,0.f,0.f,0.f,0.f,0.f,0.f};
        #pragma unroll
        for (int t = 0; t < 4; ++t) acc[t] = z8;
    }

    const int ldr = tid >> 3;           // 0..31
    const int ldc = (tid & 7) * 4;      // 0,4,...,28

    for (int k0 = 0; k0 < INPUT_DIM; k0 += KC) {
        // ---- stage x tile: 32 rows x 32 cols, one float4 per thread ----
        {
            const float4 v = *(const float4*)(x + (size_t)(bm + ldr) * INPUT_DIM + k0 + ldc);
            *(float4*)(&xs[ldr * XS_STRIDE + ldc]) = v;
        }
        // ---- stage W tile: 256 rows x 32 cols (row 255 = zeros) ----
        #pragma unroll
        for (int p = 0; p < 8; ++p) {
            const int row = ldr + p * 32;
            float4 v = make_float4(0.f, 0.f, 0.f, 0.f);
            if (row < NUM_NODES)
                v = *(const float4*)(wts + (size_t)row * INPUT_DIM + k0 + ldc);
            *(float4*)(&wl[row * WS_STRIDE + ldc]) = v;
        }
        __syncthreads();

        // prefetch next chunk while we compute (global_prefetch_b8)
        if (k0 + KC < INPUT_DIM) {
            __builtin_prefetch(x + (size_t)(bm + ldr) * INPUT_DIM + (k0 + KC) + ldc, 0, 3);
            #pragma unroll
            for (int p = 0; p < 8; p += 2)
                __builtin_prefetch(wts + (size_t)(ldr + p * 32) * INPUT_DIM + (k0 + KC) + ldc, 0, 3);
        }

        // ---- 8 k4-steps of V_WMMA_F32_16X16X4_F32 ----
        #pragma unroll
        for (int kk = 0; kk < KC; kk += 4) {
            // A(m,k): vgpr=k&1, lane = m + 16*(k>=2)  -> lane reads k pair
            v2f a = *(const v2f*)(&xs[(mw * 16 + lmod) * XS_STRIDE + kk + 2 * lhalf]);
            #pragma unroll
            for (int t = 0; t < 4; ++t) {
                const int n = (nq * 4 + t) * 16 + lmod;
                v2f b = *(const v2f*)(&wl[n * WS_STRIDE + kk + 2 * lhalf]);
                acc[t] = __builtin_amdgcn_wmma_f32_16x16x4_f32(
                    false, a, false, b, (short)0, acc[t], false, false);
            }
        }
        __syncthreads();
    }

    // ---- spill logits to LDS (reusing GEMM staging space) ----
    // C/D layout: vgpr r -> M = r (lanes 0-15) / r+8 (lanes 16-31), N = lane&15
    #pragma unroll
    for (int t = 0; t < 4; ++t) {
        #pragma unroll
        for (int r = 0; r < 8; ++r) {
            const int m = mw * 16 + r + 8 * lhalf;
            const int n = (nq * 4 + t) * 16 + lmod;
            lg[m * LG_STRIDE + n] = acc[t][r];
        }
    }
    __syncthreads();

    // ---- phase 2: one wave per sample, 4 samples per wave ----
    for (int s8 = 0; s8 < 4; ++s8) {
        const int s = wave * 4 + s8;

        // ratio = e^{clip(z)}, (1-p) = 1/(1+e); Q = prod(1-p)
        float q = 1.0f;
        #pragma unroll
        for (int j = 0; j < 8; ++j) {
            const int node = lane + 32 * j;
            float rt = 1.0f, om = 1.0f;
            if (node < NUM_NODES) {
                float z = lg[s * LG_STRIDE + node] + bias[node];
                z = fminf(10.0f, fmaxf(-10.0f, z));
                const float e = __expf(z);
                rt = e;
                om = 1.0f / (1.0f + e);
            }
            q *= om;
            lg[s * LG_STRIDE + node] = rt;   // in-place ratio table
        }
        __builtin_amdgcn_wave_barrier();

        // wave product-reduce Q
        #pragma unroll
        for (int off = 16; off; off >>= 1)
            q *= __shfl_xor(q, off, 32);

        // leaves: p_leaf = Q * prod_{level, bit=1} ratio[node_on_path]
        float wsum = 0.0f, vsum = 0.0f;
        #pragma unroll
        for (int j = 0; j < 8; ++j) {
            const int leaf = lane + 32 * j;
            float prod = q;
            #pragma unroll
            for (int lev = 0; lev < DEPTH; ++lev) {
                const int bit  = (leaf >> (DEPTH - 1 - lev)) & 1;
                const int node = (1 << lev) - 1 + (leaf >> (DEPTH - lev));
                const float rt = lg[s * LG_STRIDE + node];
                prod *= bit ? rt : 1.0f;
            }
            prod += 1e-8f;
            wsum += prod;
            vsum += prod * leafv[leaf];
        }
        #pragma unroll
        for (int off = 16; off; off >>= 1) {
            wsum += __shfl_xor(wsum, off, 32);
            vsum += __shfl_xor(vsum, off, 32);
        }
        if (lane == 0) out[bm + s] = vsum / wsum;
    }
}

extern "C" void kernel_launch(void* const* d_in, const int* in_sizes, int n_in,
                              void* d_out, int out_size, void* d_ws, size_t ws_size,
                              hipStream_t stream) {
    const float* x     = (const float*)d_in[0];
    const float* wts   = (const float*)d_in[1];
    const float* bias  = (const float*)d_in[2];
    const float* leafv = (const float*)d_in[3];
    // d_in[4] = path_mask: encoded analytically in the tree walk, not needed.
    float* out = (float*)d_out;

    dim3 grid(BATCH / MTILE);   // 256 blocks
    dim3 block(256);            // 8 waves (wave32)
    hipLaunchKernelGGL(sdt_fused_kernel, grid, block, 0, stream,
                       x, wts, bias, leafv, out);
}